// LeibnizCoupling_27943057228068
// MI455X (gfx1250) — compile-verified
//
#include <hip/hip_runtime.h>
#include <hip/hip_bf16.h>
#include <stdint.h>

typedef __attribute__((ext_vector_type(2))) float v2f;
typedef __attribute__((ext_vector_type(8))) float v8f;

#define FD 64
#define NN 10000
#define NE 160000
#define SCALE 0.125f   // F_DIM^-0.5

// tiles-per-path = 625*chan, chan = {1,1,1,3,9,3,3,3,9,3,9,9}
__constant__ int kTileCum[13] = {0,625,1250,1875,3750,9375,11250,13125,
                                 15000,20625,22500,28125,33750};
// cumulative node-level rows (N * cum-chan)
__constant__ int kRowBase[12] = {0,10000,20000,30000,60000,150000,180000,
                                 210000,240000,330000,360000,450000};

// ---------------- Phase 1: node-level linears via fp32 WMMA ----------------
// T[p] = X_p @ W[p]^T, X_p = flat [rows,64] view of h_{l_in(p)}.
// One wave -> one 16-row x 64-col output tile. 16 k-steps of K=4.
__global__ __launch_bounds__(256) void leibniz_linear_wmma(
    const float* __restrict__ h0, const float* __restrict__ h1,
    const float* __restrict__ h2, const float* __restrict__ W,
    float* __restrict__ T)
{
  const int wid  = (blockIdx.x * blockDim.x + threadIdx.x) >> 5;
  const int lane = threadIdx.x & 31;
  if (wid >= 33750) return;                 // wave-uniform: EXEC stays all-1s

  int p = 0;
  #pragma unroll
  for (int q = 0; q < 11; ++q) if (wid >= kTileCum[q + 1]) p = q + 1;
  const int t = wid - kTileCum[p];

  const float* X;
  if (p <= 2)                                    X = h0;  // l_in = 0
  else if (p == 4 || p == 8 || p == 10 || p == 11) X = h2; // l_in = 2
  else                                           X = h1;  // l_in = 1

  const float* Wp = W + (size_t)p * FD * FD;
  float*       Tp = T + ((size_t)kRowBase[p] << 6);

  const int row0 = t * 16;
  const int r    = lane & 15;     // M (for A) / N (for B,C)
  const int hh   = lane >> 4;     // half-wave select

  v8f a0 = {}, a1 = {}, a2 = {}, a3 = {};
  // A lane L, elem v: X[row0 + L%16][ks*4 + 2*(L/16) + v]
  const float* xr = X  + (size_t)(row0 + r) * FD + 2 * hh;
  // B lane L, elem v: W^T[k][n] = W[n][k] -> Wp[(nt*16 + L%16)*64 + ks*4 + 2*(L/16) + v]
  const float* w0 = Wp + (size_t)( 0 + r) * FD + 2 * hh;
  const float* w1 = Wp + (size_t)(16 + r) * FD + 2 * hh;
  const float* w2 = Wp + (size_t)(32 + r) * FD + 2 * hh;
  const float* w3 = Wp + (size_t)(48 + r) * FD + 2 * hh;

  #pragma unroll
  for (int ks = 0; ks < 16; ++ks) {
    v2f a  = *(const v2f*)(xr + ks * 4);
    v2f b0 = *(const v2f*)(w0 + ks * 4);
    v2f b1 = *(const v2f*)(w1 + ks * 4);
    v2f b2 = *(const v2f*)(w2 + ks * 4);
    v2f b3 = *(const v2f*)(w3 + ks * 4);
    a0 = __builtin_amdgcn_wmma_f32_16x16x4_f32(false, a, false, b0, (short)0, a0, false, false);
    a1 = __builtin_amdgcn_wmma_f32_16x16x4_f32(false, a, false, b1, (short)0, a1, false, false);
    a2 = __builtin_amdgcn_wmma_f32_16x16x4_f32(false, a, false, b2, (short)0, a2, false, false);
    a3 = __builtin_amdgcn_wmma_f32_16x16x4_f32(false, a, false, b3, (short)0, a3, false, false);
  }

  // C/D: VGPR rr, lane L -> (M = rr + 8*(L/16), N = L%16)
  #pragma unroll
  for (int rr = 0; rr < 8; ++rr) {
    float* dst = Tp + ((size_t)(row0 + rr + 8 * hh) << 6) + r;
    dst[0]  = a0[rr];
    dst[16] = a1[rr];
    dst[32] = a2[rr];
    dst[48] = a3[rr];
  }
}

// ---------------- Phase 2: per-(edge,feature) combine (streaming) ----------
__global__ __launch_bounds__(256) void leibniz_combine(
    const float* __restrict__ b0, const float* __restrict__ b1,
    const float* __restrict__ b2, const int* __restrict__ ei,
    const float* __restrict__ T,
    float* __restrict__ o0, float* __restrict__ o1, float* __restrict__ o2)
{
  const int gid = blockIdx.x * blockDim.x + threadIdx.x;
  if (gid >= NE * FD) return;
  const int e = gid >> 6;
  const int f = gid & 63;
  const int s = ei[e];                 // src = edge_index[0][e] (wave-uniform)

  // edge streams: non-temporal (read once, keep L2 for T)
  const float b0v = __builtin_nontemporal_load(b0 + ((size_t)e << 6) + f);
  float b1v[3];
  #pragma unroll
  for (int i = 0; i < 3; ++i)
    b1v[i] = __builtin_nontemporal_load(b1 + (((size_t)e * 3 + i) << 6) + f);
  float b2v[3][3];
  #pragma unroll
  for (int i = 0; i < 3; ++i)
    #pragma unroll
    for (int j = 0; j < 3; ++j)
      b2v[i][j] = __builtin_nontemporal_load(b2 + (((size_t)e * 9 + i * 3 + j) << 6) + f);

  // node-level transformed features: L2-resident gathers
  #define TG(base, row) T[(((size_t)(base) + (size_t)(row)) << 6) + f]
  const float t0 = TG(0,     s);
  const float t1 = TG(10000, s);
  const float t2 = TG(20000, s);
  float t3[3], t5[3], t6[3], t7[3], t9[3];
  #pragma unroll
  for (int i = 0; i < 3; ++i) {
    t3[i] = TG(30000,  s * 3 + i);
    t5[i] = TG(150000, s * 3 + i);
    t6[i] = TG(180000, s * 3 + i);
    t7[i] = TG(210000, s * 3 + i);
    t9[i] = TG(330000, s * 3 + i);
  }
  float t4[3][3], t8[3][3], t10[3][3], t11[3][3];
  #pragma unroll
  for (int i = 0; i < 3; ++i)
    #pragma unroll
    for (int j = 0; j < 3; ++j) {
      t4 [i][j] = TG(60000,  s * 9 + i * 3 + j);
      t8 [i][j] = TG(240000, s * 9 + i * 3 + j);
      t10[i][j] = TG(360000, s * 9 + i * 3 + j);
      t11[i][j] = TG(450000, s * 9 + i * 3 + j);
    }
  #undef TG

  // l_out = 0: prod(0,0) + dot(1,1) + double_dot(2,2)
  float m0 = t0 * b0v;
  #pragma unroll
  for (int i = 0; i < 3; ++i) m0 += t5[i] * b1v[i];
  #pragma unroll
  for (int i = 0; i < 3; ++i)
    #pragma unroll
    for (int j = 0; j < 3; ++j) m0 += t10[i][j] * b2v[i][j];

  // l_out = 1: prod(0,1) + prod(1,0) + cross(1,1) + mat_vec(2,1) + vec_mat(1,2)
  float m1[3];
  m1[0] = t1 * b1v[0] + t3[0] * b0v + (t6[1] * b1v[2] - t6[2] * b1v[1]);
  m1[1] = t1 * b1v[1] + t3[1] * b0v + (t6[2] * b1v[0] - t6[0] * b1v[2]);
  m1[2] = t1 * b1v[2] + t3[2] * b0v + (t6[0] * b1v[1] - t6[1] * b1v[0]);
  #pragma unroll
  for (int i = 0; i < 3; ++i)
    #pragma unroll
    for (int j = 0; j < 3; ++j) m1[i] += t8[i][j] * b1v[j];     // mat_vec
  #pragma unroll
  for (int j = 0; j < 3; ++j)
    #pragma unroll
    for (int k = 0; k < 3; ++k) m1[j] += t9[k] * b2v[k][j];     // vec_mat

  // l_out = 2: prod(0,2) + prod(2,0) + traceless outer(1,1) + traceless sym matmul(2,2)
  const float tr7 = t7[0] * b1v[0] + t7[1] * b1v[1] + t7[2] * b1v[2];
  float raw[3][3];
  #pragma unroll
  for (int i = 0; i < 3; ++i)
    #pragma unroll
    for (int j = 0; j < 3; ++j)
      raw[i][j] = t11[i][0] * b2v[0][j] + t11[i][1] * b2v[1][j] + t11[i][2] * b2v[2][j];
  const float trs  = raw[0][0] + raw[1][1] + raw[2][2]; // tr(sym) == tr(raw)
  const float diag = (1.0f / 3.0f) * (tr7 + trs);
  float m2[3][3];
  #pragma unroll
  for (int i = 0; i < 3; ++i)
    #pragma unroll
    for (int j = 0; j < 3; ++j)
      m2[i][j] = t2 * b2v[i][j] + t4[i][j] * b0v + t7[i] * b1v[j]
               + 0.5f * (raw[i][j] + raw[j][i]);
  m2[0][0] -= diag; m2[1][1] -= diag; m2[2][2] -= diag;

  // streaming stores (NT: never re-read)
  __builtin_nontemporal_store(m0 * SCALE, o0 + ((size_t)e << 6) + f);
  #pragma unroll
  for (int i = 0; i < 3; ++i)
    __builtin_nontemporal_store(m1[i] * SCALE, o1 + (((size_t)e * 3 + i) << 6) + f);
  #pragma unroll
  for (int i = 0; i < 3; ++i)
    #pragma unroll
    for (int j = 0; j < 3; ++j)
      __builtin_nontemporal_store(m2[i][j] * SCALE,
                                  o2 + (((size_t)e * 9 + i * 3 + j) << 6) + f);
}

extern "C" void kernel_launch(void* const* d_in, const int* in_sizes, int n_in,
                              void* d_out, int out_size, void* d_ws, size_t ws_size,
                              hipStream_t stream) {
  (void)in_sizes; (void)n_in; (void)out_size; (void)ws_size;
  const float* h0 = (const float*)d_in[0];
  const float* h1 = (const float*)d_in[1];
  const float* h2 = (const float*)d_in[2];
  const float* b0 = (const float*)d_in[3];
  const float* b1 = (const float*)d_in[4];
  const float* b2 = (const float*)d_in[5];
  const float* W  = (const float*)d_in[6];
  const int*   ei = (const int*)  d_in[7];

  float* T   = (float*)d_ws;                     // 540000 * 64 floats = 138.24 MB
  float* out = (float*)d_out;
  float* o0  = out;
  float* o1  = out + (size_t)NE * FD;            // msg1 after msg0
  float* o2  = out + (size_t)NE * FD * 4;        // msg2 after msg0+msg1

  // Phase 1: 33750 wave-tiles, 8 waves/block
  const int threads1 = 33750 * 32;
  leibniz_linear_wmma<<<(threads1 + 255) / 256, 256, 0, stream>>>(h0, h1, h2, W, T);

  // Phase 2: one thread per (edge, feature)
  const int threads2 = NE * FD;
  leibniz_combine<<<(threads2 + 255) / 256, 256, 0, stream>>>(b0, b1, b2, ei, T,
                                                              o0, o1, o2);
}